// ExpertGroup_48610439856255
// MI455X (gfx1250) — compile-verified
//
#include <hip/hip_runtime.h>

// ---------------------------------------------------------------------------
// MoE grouped expert FFN for MI455X (gfx1250, wave32, WMMA + TDM).
//   Kernel 1: h = gelu_tanh(x @ w1[e])   -> workspace (N x I fp32)
//   Kernel 2: y = h @ w2[e]              -> d_out     (N x H fp32)
// 128x128-tile WMMA GEMM (V_WMMA_F32_16X16X4_F32). K-chunks are staged into
// double-buffered LDS by the Tensor Data Mover (tensor_load_to_lds, tracked
// by TENSORcnt), so tile staging runs on the DMA engine while the matrix
// pipe computes. The A tile uses TDM's pad feature (pad 4 dwords every 16)
// to produce a bank-decorrelated stride-20 LDS layout for free.
// ---------------------------------------------------------------------------

typedef __attribute__((ext_vector_type(2))) float v2f;
typedef __attribute__((ext_vector_type(8))) float v8f;
typedef __attribute__((ext_vector_type(4))) unsigned u32x4;
typedef __attribute__((ext_vector_type(8))) unsigned u32x8;

#define TILE_M 128
#define TILE_N 128
#define KC     16          // K-chunk staged in LDS per iteration
#define A_STRIDE 20        // dword stride of A rows in LDS (16 data + 4 TDM pad);
                           // gcd(20,64)=4 spreads the 16 fragment rows over banks.
#define NEXPERTS 8

__device__ __forceinline__ float gelu_tanh(float x) {
  float t = 0.7978845608028654f * (x + 0.044715f * x * x * x);
  return 0.5f * x * (1.0f + tanhf(t));
}

__device__ __forceinline__ unsigned lds_addr32(const void* p) {
  // Shared-memory flat pointers keep the LDS byte offset in the low 32 bits
  // (the shared aperture lives entirely in addr[63:32]).
  return (unsigned)(unsigned long long)p;
}

// --- Tensor DMA descriptor helpers (D# groups per CDNA5 ISA ch.8) -----------
// Group 0 (4 SGPRs): [1:0]=count=1, [63:32]=lds_addr, [120:64]=global byte
// address, [127:126]=type=2.
__device__ __forceinline__ u32x4 tdm_group0(unsigned lds_addr, const void* gaddr) {
  unsigned long long ga = (unsigned long long)gaddr;
  u32x4 g;
  g.x = 1u;                                   // count=1, user descriptor
  g.y = lds_addr;                             // LDS byte address
  g.z = (unsigned)ga;                         // global_addr[31:0]
  g.w = (unsigned)(ga >> 32) | 0x80000000u;   // global_addr[56:32] | type=2
  return g;
}

// Group 1 (8 SGPRs): wg_mask=0, data_size, pad cfg, tensor/tile dims, stride0.
__device__ __forceinline__ u32x8 tdm_group1(unsigned flags,
                                            unsigned td0, unsigned td1,
                                            unsigned tile0, unsigned tile1,
                                            unsigned long long stride0) {
  u32x8 g;
  g.s0 = flags;                                   // [15:0]=0 | data_size | pads
  g.s1 = (td0 & 0xFFFFu) << 16;                   // tensor_dim0[15:0] @ [63:48]
  g.s2 = (td0 >> 16) | ((td1 & 0xFFFFu) << 16);   // td0 hi | tensor_dim1 lo
  g.s3 = (td1 >> 16) | (tile0 << 16);             // td1 hi | tile_dim0 @ [127:112]
  g.s4 = tile1 & 0xFFFFu;                         // tile_dim1; tile_dim2=0
  g.s5 = (unsigned)stride0;                       // tensor_dim0_stride[31:0]
  g.s6 = (unsigned)(stride0 >> 32) & 0xFFFFu;     // stride0[47:32]; stride1=0
  g.s7 = 0u;
  return g;
}

__device__ __forceinline__ void tdm_load(u32x4 g0, u32x8 g1) {
  // TDM DMA: global tile -> LDS. Tracked by TENSORcnt.
  asm volatile("tensor_load_to_lds %0, %1" :: "s"(g0), "s"(g1) : "memory");
}

// data_size=4B (2<<16); A adds pad_enable + pad every 16 dwords by 4 dwords.
#define TDM_FLAGS_B (2u << 16)
#define TDM_FLAGS_A (TDM_FLAGS_B | (1u << 20) | (3u << 22) | (3u << 25))

template <bool GELU>
__global__ __launch_bounds__(256)
void moe_wmma_gemm(const float* __restrict__ A,   // [Ntok, K] row-major
                   const float* __restrict__ B,   // [E][K, NC] row-major per expert
                   float* __restrict__ C,         // [Ntok, NC] row-major
                   const int* __restrict__ counts,
                   int K, int NC)
{
  __shared__ float lsA[2][TILE_M * A_STRIDE];
  __shared__ float lsB[2][KC * TILE_N];

  const int tid  = threadIdx.x;
  const int lane = tid & 31;
  const int wid  = tid >> 5;          // 8 waves
  const int wave_m = wid >> 1;        // 0..3 -> 32-row strip
  const int wave_n = wid & 1;         // 0..1 -> 64-col strip
  const int mlan   = lane & 15;
  const int khalf  = (lane >> 4) << 1; // 0 or 2 (K-pair select for f32 WMMA frags)

  const int row0 = blockIdx.x * TILE_M;
  const int col0 = blockIdx.y * TILE_N;

  // Expert for this token tile (tokens sorted by expert; tiles don't straddle).
  int e = 0;
  {
    int acc = 0;
    #pragma unroll
    for (int i = 0; i < NEXPERTS; ++i) {
      acc += counts[i];
      if (row0 < acc) { e = i; break; }
    }
  }
  const float* __restrict__ Be = B + (size_t)e * (size_t)K * (size_t)NC;

  // TDM issues are EXEC-independent scalar ops: gate on a *scalar* branch.
  const int swid = __builtin_amdgcn_readfirstlane(wid);

  unsigned ldsAbase[2], ldsBbase[2];
  #pragma unroll
  for (int p = 0; p < 2; ++p) {
    ldsAbase[p] = lds_addr32(&lsA[p][0]);
    ldsBbase[p] = lds_addr32(&lsB[p][0]);
  }

  // Invariant descriptor group-1s (tiles always fully in-bounds -> tensor
  // dims == tile dims; stride = row pitch of the source matrix).
  const u32x8 g1A = tdm_group1(TDM_FLAGS_A, KC, TILE_M, KC, TILE_M,
                               (unsigned long long)K);
  const u32x8 g1B = tdm_group1(TDM_FLAGS_B, TILE_N, KC, TILE_N, KC,
                               (unsigned long long)NC);

  auto issue_chunk = [&](int p, int kk) {
    if (swid == 0) {        // wave 0: DMA the A tile (128 rows x 16 k, padded)
      tdm_load(tdm_group0(ldsAbase[p],
                          A + (size_t)row0 * (size_t)K + (size_t)kk), g1A);
    } else if (swid == 1) { // wave 1: DMA the B tile (16 k x 128 cols, dense)
      tdm_load(tdm_group0(ldsBbase[p],
                          Be + (size_t)kk * (size_t)NC + (size_t)col0), g1B);
    }
  };

  // accumulators: 2 M-subtiles x 4 N-subtiles of 16x16
  v8f accv[2][4];
  #pragma unroll
  for (int ms = 0; ms < 2; ++ms)
    #pragma unroll
    for (int ns = 0; ns < 4; ++ns)
      accv[ms][ns] = v8f{0.f, 0.f, 0.f, 0.f, 0.f, 0.f, 0.f, 0.f};

  auto compute = [&](int p) {
    #pragma unroll
    for (int ks = 0; ks < KC / 4; ++ks) {
      const int kb = ks * 4 + khalf;   // this lane's K pair within the chunk
      v2f afrag[2];
      v2f bfrag[4];
      #pragma unroll
      for (int ms = 0; ms < 2; ++ms) {
        int row = wave_m * 32 + ms * 16 + mlan;
        afrag[ms] = *(const v2f*)&lsA[p][row * A_STRIDE + kb];
      }
      #pragma unroll
      for (int ns = 0; ns < 4; ++ns) {
        int col = wave_n * 64 + ns * 16 + mlan;
        bfrag[ns].x = lsB[p][kb * TILE_N + col];
        bfrag[ns].y = lsB[p][(kb + 1) * TILE_N + col];
      }
      #pragma unroll
      for (int ms = 0; ms < 2; ++ms)
        #pragma unroll
        for (int ns = 0; ns < 4; ++ns)
          accv[ms][ns] = __builtin_amdgcn_wmma_f32_16x16x4_f32(
              /*neg_a=*/false, afrag[ms],
              /*neg_b=*/false, bfrag[ns],
              /*c_mod=*/(short)0, accv[ms][ns],
              /*reuse_a=*/false, /*reuse_b=*/false);
    }
  };

  // --- main loop: TDM double buffering, one barrier per iteration ----------
  const int nchunks = K / KC;
  issue_chunk(0, 0);
  if (swid < 2) __builtin_amdgcn_s_wait_tensorcnt(0);
  __syncthreads();

  int p = 0;
  for (int c = 0; c < nchunks; ++c) {
    if (c + 1 < nchunks) issue_chunk(p ^ 1, (c + 1) * KC); // DMA next chunk
    compute(p);                                            // WMMA on current
    if (swid < 2) __builtin_amdgcn_s_wait_tensorcnt(0);    // DMA complete
    __syncthreads();                                       // publish to WG
    p ^= 1;
  }

  // --- epilogue: C/D layout -> global, optional GELU ---
  const int rhalf = (lane >> 4) << 3;  // +8 rows for upper half-wave
  #pragma unroll
  for (int ms = 0; ms < 2; ++ms) {
    #pragma unroll
    for (int ns = 0; ns < 4; ++ns) {
      #pragma unroll
      for (int r = 0; r < 8; ++r) {
        int grow = row0 + wave_m * 32 + ms * 16 + rhalf + r;
        int gcol = col0 + wave_n * 64 + ns * 16 + mlan;
        float v = accv[ms][ns][r];
        if (GELU) v = gelu_tanh(v);
        C[(size_t)grow * (size_t)NC + (size_t)gcol] = v;
      }
    }
  }
}

extern "C" void kernel_launch(void* const* d_in, const int* in_sizes, int n_in,
                              void* d_out, int out_size, void* d_ws, size_t ws_size,
                              hipStream_t stream) {
  const float* x_sorted = (const float*)d_in[0];   // [N, H]
  const float* w1       = (const float*)d_in[1];   // [E, H, I]
  const float* w2       = (const float*)d_in[2];   // [E, I, H]
  const int*   counts   = (const int*)d_in[3];     // [E]

  const int H = 1024;
  const int I = 4096;
  const int N = in_sizes[0] / H;   // 16384

  float* hbuf = (float*)d_ws;      // N * I fp32 intermediate (256 MB)

  dim3 blk(256);
  dim3 g1(N / TILE_M, I / TILE_N);
  moe_wmma_gemm<true><<<g1, blk, 0, stream>>>(x_sorted, w1, hbuf, counts, H, I);

  dim3 g2(N / TILE_M, H / TILE_N);
  moe_wmma_gemm<false><<<g2, blk, 0, stream>>>(hbuf, w2, (float*)d_out, counts, I, H);
}